// Net_4105988735294
// MI455X (gfx1250) — compile-verified
//
#include <hip/hip_runtime.h>
#include <math.h>

// ---------------- constants matching the reference ----------------
#define BATCH     2
#define CIN       256
#define A_TOTAL   159882          // 3*(200^2+100^2+50^2+25^2+13^2)
#define K_TOT     4507            // 1000+1000+1000+1000+507
#define POST_NMS  1000
#define IMG_F     800.0f
#define NMS_TH    0.7f
#define MIN_SIZE  1e-3f
#define BBOX_CLIP 4.135166556742356f   // log(1000/16)

typedef __attribute__((ext_vector_type(2))) float v2f;
typedef __attribute__((ext_vector_type(8))) float v8f;

__device__ inline v8f wmma_f32_4(v2f a, v2f b, v8f c) {
    // V_WMMA_F32_16X16X4_F32 : D(16x16 f32) = A(16x4) * B(4x16) + C
    return __builtin_amdgcn_wmma_f32_16x16x4_f32(
        false, a, false, b, (short)0, c, false, false);
}

__device__ inline unsigned fkey(float f) {
    unsigned u = __float_as_uint(f);
    return (u & 0x80000000u) ? ~u : (u | 0x80000000u);
}

// =================================================================
// Kernel 0: repack conv weights [co][ci][3][3] -> wpk[tap][ci/2][co][2]
// so an A fragment (pair of consecutive ci for one co) is one b64 load,
// coalesced across lanes (co contiguous).
// =================================================================
__global__ void repack_w_kernel(const float* __restrict__ w, float* __restrict__ wpk)
{
    const int g = blockIdx.x * blockDim.x + threadIdx.x;
    if (g >= 9 * 128 * 256 * 2) return;
    const int lo   = g & 1;
    const int co   = (g >> 1) & 255;
    const int rest = g >> 9;
    const int ci2  = rest & 127;
    const int tap  = rest >> 7;
    wpk[g] = w[((size_t)co * CIN + 2 * ci2 + lo) * 9 + tap];
}

// =================================================================
// Kernel 1: fused conv3x3(256->256)+bias+ReLU -> cls(3) & box(12) heads
// One block = 2 rows x 16 cols (32 pixels). ci processed in two halves of
// 128 (halo re-staged between halves, accumulators persist) to stay in LDS.
// 8 waves; wave w owns co [32w,32w+32); 4 accumulators = 2 co-tiles x 2 rows.
// Per K-step: 2x global_load_b64 (packed weights) + 2x ds_load_b64 + 4 WMMA.
// =================================================================
union SMem {
    unsigned long long _align8;
    float hal[64][144];    // [ci_pair(128 ci half)][(4*18 halo px)*2 + ci&1] 36864 B
    float t[CIN][33];      // ReLU'd activation tile 32 px (overlapped)      33792 B
};

__global__ __launch_bounds__(256)
void rpn_fused_kernel(const float* __restrict__ feat,
                      const float* __restrict__ wpk,  const float* __restrict__ conv_b,
                      const float* __restrict__ cls_w, const float* __restrict__ cls_b,
                      const float* __restrict__ box_w, const float* __restrict__ box_b,
                      float* __restrict__ obj, float* __restrict__ del,
                      int H, int TX, int aoff)
{
    const int b   = blockIdx.y;
    const int y0  = (blockIdx.x / TX) * 2;        // two output rows: y0, y0+1
    const int x0  = (blockIdx.x % TX) * 16;
    const int P   = H * H;
    const int tid = threadIdx.x;
    const int wave = tid >> 5;
    const int lane = tid & 31;
    const int half = lane >> 4;     // 0: K pair {0,1}; 1: K pair {2,3}
    const int m    = lane & 15;     // A: co-row in tile / B: pixel column

    __shared__ SMem sm;

    const int co0 = wave * 32;
    const int co1 = co0 + 16;
    v8f acc00 = {}, acc01 = {}, acc10 = {}, acc11 = {};

    for (int ch = 0; ch < 2; ++ch) {              // ci half: [128*ch, 128*ch+128)
        // ---- stage halo: rows y0-1..y0+2, cols x0-1..x0+16, 128 ci ----
        const float* fb = feat + ((size_t)b * CIN + ch * 128) * P;
        for (int e = tid; e < 72 * 128; e += 256) {
            const int ci = e / 72;                // local 0..127
            const int hp = e % 72;                // 4 rows x 18 cols
            const int r  = hp / 18, c = hp % 18;
            const int iy = y0 + r - 1, ix = x0 + c - 1;
            float v = 0.0f;
            if (iy >= 0 && iy < H && ix >= 0 && ix < H)
                v = fb[(size_t)ci * P + (size_t)iy * H + ix];
            sm.hal[ci >> 1][hp * 2 + (ci & 1)] = v;
        }
        __syncthreads();

        for (int tap = 0; tap < 9; ++tap) {
            const int kh = tap / 3, kw = tap % 3;             // 0..2
            const int hp0 = kh * 18 + m + kw;                 // row-tile 0 B pixel
            const int hp1 = hp0 + 18;                         // row-tile 1 B pixel
            const v2f* wt = (const v2f*)wpk
                          + ((size_t)tap * 128 + (size_t)ch * 64) * 256;
            if (tap < 8)
                __builtin_prefetch(wpk + (((size_t)(tap + 1) * 128 + (size_t)ch * 64) * 256) * 2
                                       + tid * 64, 0, 1);

#pragma unroll 4
            for (int ci0 = 0; ci0 < 128; ci0 += 4) {
                const int pr = (ci0 >> 1) + half;             // local ci pair 0..63
                const v2f b0 = *(const v2f*)&sm.hal[pr][hp0 * 2];
                const v2f b1 = *(const v2f*)&sm.hal[pr][hp1 * 2];
                const v2f a0 = wt[pr * 256 + co0 + m];
                const v2f a1 = wt[pr * 256 + co1 + m];
                acc00 = wmma_f32_4(a0, b0, acc00);
                acc01 = wmma_f32_4(a0, b1, acc01);
                acc10 = wmma_f32_4(a1, b0, acc10);
                acc11 = wmma_f32_4(a1, b1, acc11);
            }
        }
        __syncthreads();   // all halo reads done before re-staging / union reuse
    }

    // C layout: VGPR r -> rows (r, r+8) split by lane half; col = lane&15
    // pixel index in tile: row-tile0 -> m, row-tile1 -> 16+m
    for (int r = 0; r < 8; ++r) {
        int co = co0 + r + 8 * half;
        sm.t[co][m]      = fmaxf(acc00[r] + conv_b[co], 0.0f);
        sm.t[co][16 + m] = fmaxf(acc01[r] + conv_b[co], 0.0f);
        co = co1 + r + 8 * half;
        sm.t[co][m]      = fmaxf(acc10[r] + conv_b[co], 0.0f);
        sm.t[co][16 + m] = fmaxf(acc11[r] + conv_b[co], 0.0f);
    }
    __syncthreads();

    // 1x1 heads: 32 px * (3 cls + 12 box) = 480 dot products of length 256
    for (int e = tid; e < 480; e += 256) {
        const int pxl = e / 15;                   // 0..31
        const int o   = e % 15;
        const int rr  = pxl >> 4, cc = pxl & 15;
        if (x0 + cc < H && y0 + rr < H) {
            const int p2 = (y0 + rr) * H + x0 + cc;
            const float* wv; float acc;
            if (o < 3) { wv = cls_w + (size_t)o * CIN;       acc = cls_b[o];     }
            else       { wv = box_w + (size_t)(o - 3) * CIN; acc = box_b[o - 3]; }
            for (int ci = 0; ci < CIN; ++ci) acc += wv[ci] * sm.t[ci][pxl];
            if (o < 3)
                obj[(size_t)b * A_TOTAL + aoff + (size_t)p2 * 3 + o] = acc;
            else
                del[((size_t)b * A_TOTAL + aoff) * 4 + (size_t)p2 * 12 + (o - 3)] = acc;
        }
    }
}

// =================================================================
// Kernel 2: per-(image,level) top-k via 4-round byte radix-select
// =================================================================
__global__ __launch_bounds__(1024)
void topk_kernel(const float* __restrict__ obj, int* __restrict__ cand,
                 int n, int aoff, int k, int seg)
{
    const int b = blockIdx.x;
    const float* sc = obj + (size_t)b * A_TOTAL + aoff;
    int* out = cand + (size_t)b * K_TOT + seg;
    const int tid = threadIdx.x;

    __shared__ unsigned bins[256];
    __shared__ unsigned sh_prefix, sh_need, sh_cntG, sh_cntE;
    if (tid == 0) { sh_prefix = 0u; sh_need = (unsigned)k; }
    __syncthreads();

    for (int d = 3; d >= 0; --d) {
        if (tid < 256) bins[tid] = 0u;
        __syncthreads();
        const unsigned pre  = sh_prefix;
        const unsigned need = sh_need;
        const unsigned him  = (d == 3) ? 0u : (0xFFFFFFFFu << ((d + 1) * 8));
        for (int i = tid; i < n; i += blockDim.x) {
            unsigned key = fkey(sc[i]);
            if ((key & him) == pre)
                atomicAdd(&bins[(key >> (d * 8)) & 0xFFu], 1u);
        }
        __syncthreads();
        if (tid == 0) {
            unsigned cum = 0;
            for (int v = 255; v >= 0; --v) {
                unsigned c = bins[v];
                if (cum + c >= need) {
                    sh_need   = need - cum;
                    sh_prefix = pre | ((unsigned)v << (d * 8));
                    break;
                }
                cum += c;
            }
        }
        __syncthreads();
    }
    const unsigned T    = sh_prefix;
    const unsigned need = sh_need;             // how many == T are taken
    const unsigned nG   = (unsigned)k - need;  // count strictly greater
    if (tid == 0) { sh_cntG = 0u; sh_cntE = 0u; }
    __syncthreads();
    for (int i = tid; i < n; i += blockDim.x) {
        unsigned key = fkey(sc[i]);
        if (key > T) {
            unsigned pos = atomicAdd(&sh_cntG, 1u);
            out[pos] = aoff + i;
        } else if (key == T) {
            unsigned e = atomicAdd(&sh_cntE, 1u);
            if (e < need) out[nG + e] = aoff + i;
        }
    }
}

// =================================================================
// Kernel 3: gather + sigmoid + decode + clip + validity + level offset
// =================================================================
__global__ void gather_decode_kernel(const float* __restrict__ obj,
                                     const float* __restrict__ del,
                                     const int*   __restrict__ cand,
                                     const float* __restrict__ anchors,
                                     float* __restrict__ boxes,
                                     float* __restrict__ nbox,
                                     float* __restrict__ svalid)
{
    const int g = blockIdx.x * blockDim.x + threadIdx.x;
    if (g >= BATCH * K_TOT) return;
    const int b = g / K_TOT, s = g % K_TOT;
    const int lvl = (s >= 4000) ? 4 : (s / 1000);

    const int a = cand[g];
    const float logit = obj[(size_t)b * A_TOTAL + a];
    const float score = 1.0f / (1.0f + expf(-logit));

    const float* an = anchors + (size_t)a * 4;
    const float w  = an[2] - an[0], h  = an[3] - an[1];
    const float cx = an[0] + 0.5f * w, cy = an[1] + 0.5f * h;

    const float* dd = del + ((size_t)b * A_TOTAL + a) * 4;
    const float dx = dd[0], dy = dd[1];
    const float dw = fminf(dd[2], BBOX_CLIP);
    const float dh = fminf(dd[3], BBOX_CLIP);
    const float pcx = dx * w + cx, pcy = dy * h + cy;
    const float pw  = expf(dw) * w, ph  = expf(dh) * h;

    float x0 = fminf(fmaxf(pcx - 0.5f * pw, 0.0f), IMG_F);
    float y0 = fminf(fmaxf(pcy - 0.5f * ph, 0.0f), IMG_F);
    float x1 = fminf(fmaxf(pcx + 0.5f * pw, 0.0f), IMG_F);
    float y1 = fminf(fmaxf(pcy + 0.5f * ph, 0.0f), IMG_F);

    const bool valid = ((x1 - x0) >= MIN_SIZE) && ((y1 - y0) >= MIN_SIZE) && (score >= 0.0f);
    const float off = (float)lvl * (IMG_F + 1.0f);

    boxes[(size_t)g * 4 + 0] = x0; boxes[(size_t)g * 4 + 1] = y0;
    boxes[(size_t)g * 4 + 2] = x1; boxes[(size_t)g * 4 + 3] = y1;
    nbox[(size_t)g * 4 + 0] = x0 + off; nbox[(size_t)g * 4 + 1] = y0 + off;
    nbox[(size_t)g * 4 + 2] = x1 + off; nbox[(size_t)g * 4 + 3] = y1 + off;
    svalid[g] = valid ? score : -INFINITY;
}

// =================================================================
// Kernel 4: per-image bitonic sort (desc score, asc index) in 64KB LDS
// =================================================================
#define SORT_N 8192
__global__ __launch_bounds__(1024)
void sort_kernel(const float* __restrict__ svalid, int* __restrict__ order)
{
    const int b = blockIdx.x;
    __shared__ unsigned long long key[SORT_N];
    const int tid = threadIdx.x;

    for (int i = tid; i < SORT_N; i += 1024) {
        unsigned long long kk = 0ull;
        if (i < K_TOT) {
            unsigned sk = fkey(svalid[(size_t)b * K_TOT + i]);
            kk = ((unsigned long long)sk << 32) | (unsigned long long)(0xFFFFFFFFu - (unsigned)i);
        }
        key[i] = kk;
    }
    __syncthreads();

    for (int k2 = 2; k2 <= SORT_N; k2 <<= 1) {
        for (int j = k2 >> 1; j > 0; j >>= 1) {
            for (int i = tid; i < SORT_N; i += 1024) {
                const int ixj = i ^ j;
                if (ixj > i) {
                    const bool up = ((i & k2) == 0);
                    unsigned long long a = key[i], c = key[ixj];
                    if (up ? (a < c) : (a > c)) { key[i] = c; key[ixj] = a; }  // descending
                }
            }
            __syncthreads();
        }
    }
    for (int i = tid; i < K_TOT; i += 1024)
        order[(size_t)b * K_TOT + i] =
            (int)(0xFFFFFFFFu - (unsigned)(key[i] & 0xFFFFFFFFull));
}

// =================================================================
// Kernel 5: greedy NMS (sorted order) + compaction of top POST_NMS into d_out
// d_out layout: out_b [B,1000,4] then out_s [B,1000] (flat, zero padded)
// =================================================================
__global__ __launch_bounds__(1024)
void nms_kernel(const int* __restrict__ order,
                const float* __restrict__ nbox,
                const float* __restrict__ boxes,
                const float* __restrict__ svalid,
                float* __restrict__ sbox, float* __restrict__ sscr, int* __restrict__ sidx,
                float* __restrict__ out)
{
    const int b = blockIdx.x;
    const int tid = threadIdx.x;
    __shared__ unsigned char keep[K_TOT];

    // stage in sorted order + init keep + zero this image's output slice
    for (int r = tid; r < K_TOT; r += 1024) {
        const int s = order[(size_t)b * K_TOT + r];
        const size_t dst = ((size_t)b * K_TOT + r) * 4;
        const size_t src = ((size_t)b * K_TOT + s) * 4;
        sbox[dst + 0] = nbox[src + 0]; sbox[dst + 1] = nbox[src + 1];
        sbox[dst + 2] = nbox[src + 2]; sbox[dst + 3] = nbox[src + 3];
        sscr[(size_t)b * K_TOT + r] = svalid[(size_t)b * K_TOT + s];
        sidx[(size_t)b * K_TOT + r] = s;
        keep[r] = 1;
    }
    for (int i = tid; i < POST_NMS * 4; i += 1024) out[(size_t)b * POST_NMS * 4 + i] = 0.0f;
    for (int i = tid; i < POST_NMS;     i += 1024) out[(size_t)BATCH * POST_NMS * 4 + (size_t)b * POST_NMS + i] = 0.0f;
    __syncthreads();

    const float* sb = sbox + (size_t)b * K_TOT * 4;
    for (int i = 0; i < K_TOT - 1; ++i) {
        if (keep[i]) {
            const float bx0 = sb[(size_t)i * 4 + 0], by0 = sb[(size_t)i * 4 + 1];
            const float bx1 = sb[(size_t)i * 4 + 2], by1 = sb[(size_t)i * 4 + 3];
            const float ai = (bx1 - bx0) * (by1 - by0);
            for (int r = i + 1 + tid; r < K_TOT; r += 1024) {
                if (keep[r]) {
                    const float cx0 = sb[(size_t)r * 4 + 0], cy0 = sb[(size_t)r * 4 + 1];
                    const float cx1 = sb[(size_t)r * 4 + 2], cy1 = sb[(size_t)r * 4 + 3];
                    const float iw = fminf(bx1, cx1) - fmaxf(bx0, cx0);
                    const float ih = fminf(by1, cy1) - fmaxf(by0, cy0);
                    const float inter = fmaxf(iw, 0.0f) * fmaxf(ih, 0.0f);
                    const float ar = (cx1 - cx0) * (cy1 - cy0);
                    const float iou = inter / (ai + ar - inter);
                    if (iou > NMS_TH) keep[r] = 0;
                }
            }
        }
        __syncthreads();
    }

    if (tid == 0) {
        int cnt = 0;
        for (int r = 0; r < K_TOT && cnt < POST_NMS; ++r) {
            const float sc = sscr[(size_t)b * K_TOT + r];
            if (keep[r] && sc > -1e30f) {
                const int s = sidx[(size_t)b * K_TOT + r];
                const size_t src = ((size_t)b * K_TOT + s) * 4;
                const size_t dst = ((size_t)b * POST_NMS + cnt) * 4;
                out[dst + 0] = boxes[src + 0]; out[dst + 1] = boxes[src + 1];
                out[dst + 2] = boxes[src + 2]; out[dst + 3] = boxes[src + 3];
                out[(size_t)BATCH * POST_NMS * 4 + (size_t)b * POST_NMS + cnt] = sc;
                ++cnt;
            }
        }
    }
}

// =================================================================
extern "C" void kernel_launch(void* const* d_in, const int* in_sizes, int n_in,
                              void* d_out, int out_size, void* d_ws, size_t ws_size,
                              hipStream_t stream)
{
    (void)in_sizes; (void)n_in; (void)out_size; (void)ws_size;
    const float* feat[5] = { (const float*)d_in[0], (const float*)d_in[1],
                             (const float*)d_in[2], (const float*)d_in[3],
                             (const float*)d_in[4] };
    const float* conv_w = (const float*)d_in[5];
    const float* conv_b = (const float*)d_in[6];
    const float* cls_w  = (const float*)d_in[7];
    const float* cls_b  = (const float*)d_in[8];
    const float* box_w  = (const float*)d_in[9];
    const float* box_b  = (const float*)d_in[10];
    const float* anchors= (const float*)d_in[11];

    // workspace carving (~9.3 MB total, L2-resident)
    char* base = (char*)d_ws;
    size_t off = 0;
    auto carve = [&](size_t bytes) -> void* {
        void* p = base + off;
        off = (off + bytes + 255) & ~(size_t)255;
        return p;
    };
    float* wpk    = (float*)carve(sizeof(float) * 9 * 128 * 256 * 2);   // packed weights
    float* obj    = (float*)carve(sizeof(float) * BATCH * A_TOTAL);
    float* del    = (float*)carve(sizeof(float) * BATCH * A_TOTAL * 4);
    int*   cand   = (int*)  carve(sizeof(int)   * BATCH * K_TOT);
    float* boxes  = (float*)carve(sizeof(float) * BATCH * K_TOT * 4);
    float* nbox   = (float*)carve(sizeof(float) * BATCH * K_TOT * 4);
    float* svalid = (float*)carve(sizeof(float) * BATCH * K_TOT);
    int*   ordr   = (int*)  carve(sizeof(int)   * BATCH * K_TOT);
    float* sbox   = (float*)carve(sizeof(float) * BATCH * K_TOT * 4);
    float* sscr   = (float*)carve(sizeof(float) * BATCH * K_TOT);
    int*   sidx   = (int*)  carve(sizeof(int)   * BATCH * K_TOT);

    static const int HWs[5]  = {200, 100, 50, 25, 13};
    static const int offs[5] = {0, 120000, 150000, 157500, 159375};
    static const int ks[5]   = {1000, 1000, 1000, 1000, 507};
    static const int segs[5] = {0, 1000, 2000, 3000, 4000};

    repack_w_kernel<<<(9 * 128 * 256 * 2 + 255) / 256, 256, 0, stream>>>(conv_w, wpk);

    for (int l = 0; l < 5; ++l) {
        const int H  = HWs[l];
        const int TX = (H + 15) / 16;
        const int TY = (H + 1) / 2;
        dim3 grid(TY * TX, BATCH);
        rpn_fused_kernel<<<grid, 256, 0, stream>>>(
            feat[l], wpk, conv_b, cls_w, cls_b, box_w, box_b,
            obj, del, H, TX, offs[l]);
    }
    for (int l = 0; l < 5; ++l) {
        topk_kernel<<<BATCH, 1024, 0, stream>>>(
            obj, cand, 3 * HWs[l] * HWs[l], offs[l], ks[l], segs[l]);
    }
    gather_decode_kernel<<<(BATCH * K_TOT + 255) / 256, 256, 0, stream>>>(
        obj, del, cand, anchors, boxes, nbox, svalid);
    sort_kernel<<<BATCH, 1024, 0, stream>>>(svalid, ordr);
    nms_kernel<<<BATCH, 1024, 0, stream>>>(
        ordr, nbox, boxes, svalid, sbox, sscr, sidx, (float*)d_out);
}